// JointConditionalDistributionBlock_49735721287943
// MI455X (gfx1250) — compile-verified
//
#include <hip/hip_runtime.h>
#include <hip/hip_bf16.h>
#include <stdint.h>

// ---------------------------------------------------------------------------
// JointConditionalDistributionBlock for MI455X (gfx1250, wave32)
//
// Shapes: H_P=3, F_P=4, K=12, D=7.  NH = NF = 12^3 = 1728, NPX = 12^4.
// Memory-bound: one pass over 12^7 floats (143 MB) dominates (~6.2us @ 23.3TB/s).
// CDNA5 paths used: v_wmma_f32_16x16x4_f32 (tensor-product expansion is a
// rank-1 matmul), tensor_load_to_lds + s_wait_tensorcnt (TDM streaming of the
// big bias tensor into LDS, one descriptor per wave).
// ---------------------------------------------------------------------------

#define KB   12
#define H_P  3
#define F_P  4
#define NH   1728      // 12^3 h-triples
#define NF   1728      // 12^3 (f0,f1,f2) rows per h-slice
#define NPX  20736     // 12^4

typedef __attribute__((ext_vector_type(2)))  float    v2f;
typedef __attribute__((ext_vector_type(8)))  float    v8f;
typedef __attribute__((ext_vector_type(4)))  uint32_t v4u;
typedef __attribute__((ext_vector_type(8)))  int      v8i;
typedef __attribute__((ext_vector_type(4)))  int      v4i;

// ---------------------------------------------------------------------------
// K1: u = t0 (x) t1, v = t2 (x) t3 via V_WMMA_F32_16X16X4_F32 outer products.
// A: 16x4 with column K=0 = t_even (lanes 0-15 hold M, VGPR0=K0 / K2 split);
// B: 4x16 with row K=0 = t_odd.  D[m,n] = t_even[m]*t_odd[n].
// One wave, EXEC all ones (WMMA requirement).
// ---------------------------------------------------------------------------
__global__ __launch_bounds__(32)
void k_outer_wmma(const float* __restrict__ x, const float* __restrict__ tpx_bias,
                  float* __restrict__ u_ws, float* __restrict__ v_ws) {
    const int lane = threadIdx.x;
    float t0 = 0.f, t1 = 0.f, t2 = 0.f, t3 = 0.f;
    if (lane < KB) {
        t0 = x[0 * KB + lane] + tpx_bias[0 * KB + lane];
        t1 = x[1 * KB + lane] + tpx_bias[1 * KB + lane];
        t2 = x[2 * KB + lane] + tpx_bias[2 * KB + lane];
        t3 = x[3 * KB + lane] + tpx_bias[3 * KB + lane];
    }
    // VGPR0 of A holds K=0 on lanes 0-15 (K=2 on 16-31: zero); VGPR1 zero.
    // VGPR0 of B holds K=0 row on lanes 0-15; all other K rows zero.
    v2f a0 = {0.f, 0.f}, b0 = {0.f, 0.f}, a1 = {0.f, 0.f}, b1 = {0.f, 0.f};
    if (lane < 16) { a0.x = t0; b0.x = t1; a1.x = t2; b1.x = t3; }
    v8f cz = {};
    v8f du = __builtin_amdgcn_wmma_f32_16x16x4_f32(false, a0, false, b0,
                                                   (short)0, cz, false, false);
    v8f dv = __builtin_amdgcn_wmma_f32_16x16x4_f32(false, a1, false, b1,
                                                   (short)0, cz, false, false);
    // C/D layout: VGPR j -> row M=j (lanes 0-15) / M=j+8 (lanes 16-31), N=lane&15
    const int mbase = (lane >= 16) ? 8 : 0;
    const int n = lane & 15;
#pragma unroll
    for (int j = 0; j < 8; ++j) {
        u_ws[(mbase + j) * 16 + n] = du[j];
        v_ws[(mbase + j) * 16 + n] = dv[j];
    }
}

// ---------------------------------------------------------------------------
// K2: P_X rows.  px[(a,b,c), d] = softmax_d( u[a,b]*v[c,d] + bias_X )
// ---------------------------------------------------------------------------
__global__ __launch_bounds__(256)
void k_px(const float* __restrict__ u_ws, const float* __restrict__ v_ws,
          const float* __restrict__ bias_X, float* __restrict__ px) {
    const int row = blockIdx.x * blockDim.x + threadIdx.x;   // (a,b,c)
    if (row >= NF) return;
    const int c  = row % KB;
    const int ab = row / KB;
    const float uab = u_ws[(ab / KB) * 16 + (ab % KB)];
    float vals[KB];
    float m = -3.4e38f;
#pragma unroll
    for (int d = 0; d < KB; ++d) {
        float val = uab * v_ws[c * 16 + d] + bias_X[row * KB + d];
        vals[d] = val;
        m = fmaxf(m, val);
    }
    float z = 0.f;
#pragma unroll
    for (int d = 0; d < KB; ++d) { vals[d] = __expf(vals[d] - m); z += vals[d]; }
    const float inv = 1.f / z;
#pragma unroll
    for (int d = 0; d < KB; ++d) px[row * KB + d] = vals[d] * inv;
}

// ---------------------------------------------------------------------------
// K3: the 143 MB streaming pass.  Block b owns h-slice b (1728 rows x 12).
// Each of the 8 waves TDM-loads its contiguous 216-row (10368 B) chunk into
// its private LDS region, waits on its own TENSORcnt, then computes
//   sum_rows ( sum_d exp(b_d - max)*px_d ) / ( sum_d exp(b_d - max) )
// j_k is constant over the softmax axis so softmax(j_k+bias)=softmax(bias).
// ---------------------------------------------------------------------------
__global__ __launch_bounds__(256)
void k_mk(const float* __restrict__ biasYgX, const float* __restrict__ px,
          float* __restrict__ mk) {
    __shared__ float lds_tile[NF * KB];   // 82944 bytes (<320KB/WGP)
    __shared__ float acc;
    const int h    = blockIdx.x;
    const int tid  = threadIdx.x;
    const int wave = tid >> 5;
    const int lane = tid & 31;
    if (tid == 0) acc = 0.f;

    const int rows_per_wave = NF / 8;          // 216
    const int elems = rows_per_wave * KB;      // 2592 floats = 10368 bytes

    const uint64_t gaddr = (uint64_t)(uintptr_t)biasYgX +
        ((uint64_t)h * NF + (uint64_t)wave * rows_per_wave) * (uint64_t)(KB * 4);
    const uint32_t lds_off =
        (uint32_t)(uintptr_t)(&lds_tile[0]) + (uint32_t)(wave * elems * 4);

    // ---- D# group 0: count=1 | lds_addr | global_addr[56:0] | type=2 ----
    v4u g0 = { 1u,
               lds_off,
               (uint32_t)(gaddr & 0xffffffffu),
               (uint32_t)((gaddr >> 32) & 0x01ffffffu) | (2u << 30) };
    // ---- D# group 1: data_size=2 (4B); tensor_dim0=stride0=elems;
    //      tensor_dim1=1; tile_dim0=elems (1-D tile, tile_dim1/2 unused) ----
    v8i g1 = { (int)(2u << 16),
               (int)(((uint32_t)elems & 0xffffu) << 16),
               (int)((((uint32_t)elems >> 16) & 0xffffu) | (1u << 16)),
               (int)(((uint32_t)elems & 0xffffu) << 16),
               0,
               (int)elems,
               0, 0 };
    v4i g2 = {0, 0, 0, 0};
    v4i g3 = {0, 0, 0, 0};
    v8i g4 = {0, 0, 0, 0, 0, 0, 0, 0};   // clang-23 6-arg form: extra group
    __builtin_amdgcn_tensor_load_to_lds(g0, g1, g2, g3, g4, 0);
    __builtin_amdgcn_s_wait_tensorcnt(0);     // this wave's TDM complete
    __syncthreads();                          // acc=0 visible before adds

    float partial = 0.f;
    const float* ldsw = &lds_tile[wave * elems];
    const int f012_base = wave * rows_per_wave;
    for (int r = lane; r < rows_per_wave; r += 32) {
        const float* brow = &ldsw[r * KB];
        const float* prow = &px[(f012_base + r) * KB];
        float m = brow[0];
#pragma unroll
        for (int d = 1; d < KB; ++d) m = fmaxf(m, brow[d]);
        float z = 0.f, s = 0.f;
#pragma unroll
        for (int d = 0; d < KB; ++d) {
            float e = __expf(brow[d] - m);
            z += e;
            s += e * prow[d];
        }
        partial += s / z;
    }
#pragma unroll
    for (int off = 16; off > 0; off >>= 1)
        partial += __shfl_down(partial, off, 32);
    if (lane == 0) atomicAdd(&acc, partial);
    __syncthreads();
    if (tid == 0) mk[h] = acc;
}

// ---------------------------------------------------------------------------
// K4: marginals over each H dim + softmax -> out[3][12]
// ---------------------------------------------------------------------------
__global__ __launch_bounds__(64)
void k_final(const float* __restrict__ mk, float* __restrict__ out) {
    __shared__ float marg[H_P][KB];
    const int tid = threadIdx.x;
    if (tid < H_P * KB) {
        const int i = tid / KB, k = tid % KB;
        float s = 0.f;
        for (int a = 0; a < KB; ++a)
            for (int b = 0; b < KB; ++b) {
                int h0, h1, h2;
                if (i == 0)      { h0 = k; h1 = a; h2 = b; }
                else if (i == 1) { h0 = a; h1 = k; h2 = b; }
                else             { h0 = a; h1 = b; h2 = k; }
                s += mk[(h0 * KB + h1) * KB + h2];
            }
        marg[i][k] = s;
    }
    __syncthreads();
    if (tid < H_P * KB) {
        const int i = tid / KB, k = tid % KB;
        float m = marg[i][0];
        for (int d = 1; d < KB; ++d) m = fmaxf(m, marg[i][d]);
        float z = 0.f;
        for (int d = 0; d < KB; ++d) z += __expf(marg[i][d] - m);
        out[tid] = __expf(marg[i][k] - m) / z;
    }
}

// ---------------------------------------------------------------------------
extern "C" void kernel_launch(void* const* d_in, const int* in_sizes, int n_in,
                              void* d_out, int out_size, void* d_ws, size_t ws_size,
                              hipStream_t stream) {
    (void)in_sizes; (void)n_in; (void)out_size; (void)ws_size;
    // setup_inputs order: x, context_x, context_y, H_bandwidth, tpx_bias,
    //                     bias_Y_given_X, bias_X
    const float* x        = (const float*)d_in[0];
    const float* tpx_bias = (const float*)d_in[4];   // [1,4,12] -> flat 48
    const float* biasYgX  = (const float*)d_in[5];   // [12]^7
    const float* bias_X   = (const float*)d_in[6];   // [12]^4
    // context_x / context_y / H_bandwidth only produce a constant shift that
    // cancels inside softmax(axis=-1): mathematically irrelevant to output.

    float* ws   = (float*)d_ws;
    float* u_ws = ws;                 // 256 floats (16x16, top-left 12x12 used)
    float* v_ws = ws + 256;           // 256 floats
    float* px   = ws + 512;           // 20736 floats
    float* mk   = ws + 512 + NPX;     // 1728 floats

    k_outer_wmma<<<1, 32, 0, stream>>>(x, tpx_bias, u_ws, v_ws);
    k_px<<<(NF + 255) / 256, 256, 0, stream>>>(u_ws, v_ws, bias_X, px);
    k_mk<<<NH, 256, 0, stream>>>(biasYgX, px, mk);
    k_final<<<1, 64, 0, stream>>>(mk, (float*)d_out);
}